// sEBEncoder_670014898413
// MI455X (gfx1250) — compile-verified
//
#include <hip/hip_runtime.h>
#include <cstddef>

typedef __attribute__((ext_vector_type(2))) float v2f;
typedef __attribute__((ext_vector_type(8))) float v8f;

namespace {
constexpr int Tn = 36;
constexpr int Pn = 40;
constexpr int Kn = 4 * Pn + 1;   // 161
constexpr int Bn = 16;
constexpr int Nn = 128;
constexpr int NB = Nn * Bn;      // 2048 (n,b) pairs
constexpr int KP = 176;          // K padded to 11 WMMA tiles of 16
constexpr int NT = KP / 16;      // 11
constexpr int TP = 48;           // T padded to 3 WMMA tiles of 16
constexpr int MT = TP / 16;      // 3

// d_out offsets (floats): mu, Sigma, loss, var — concatenated in return order
constexpr size_t OFF_MU    = 0;
constexpr size_t OFF_SIGMA = (size_t)NB * Kn;                     // 329,728
constexpr size_t OFF_LOSS  = OFF_SIGMA + (size_t)NB * Kn * Kn;    // 53,415,936
constexpr size_t OFF_VAR   = OFF_LOSS + 1;

// workspace offsets (floats)
constexpr size_t WS_D   = 0;                                  // T*K = 5796
constexpr size_t WS_DTY = 8192;                               // NB*K
constexpr size_t WS_M   = WS_DTY + (size_t)NB * Kn;           // NB*T*T
constexpr size_t WS_S   = WS_M + (size_t)NB * Tn * Tn;        // NB*T*T
constexpr size_t WS_PL  = WS_S + (size_t)NB * Tn * Tn;        // NB per-pair loss
} // namespace

// ---------------------------------------------------------------- dictionary
__device__ __forceinline__ float dict_elem(int k, int i,
                                           const float* __restrict__ Drr,
                                           const float* __restrict__ Dth) {
  if (k == 0) return 1.0f;
  const int g = (k - 1) / Pn;       // 0..3
  const int p = (k - 1) % Pn;
  const float r  = Drr[p];
  const float th = Dth[p];
  float pr = powf(r, (float)i);     // r^i
  if ((g == 1 || g == 3) && (i & 1)) pr = -pr;   // (-r)^i
  const float ang = (float)i * th;
  return pr * ((g <= 1) ? cosf(ang) : sinf(ang));
}

__global__ void k_dict(const float* __restrict__ Drr, const float* __restrict__ Dth,
                       float* __restrict__ Dg) {
  const int k = threadIdx.x;
  if (k >= Kn) return;
  float nrm2 = 0.0f;
  for (int i = 0; i < Tn; ++i) { const float v = dict_elem(k, i, Drr, Dth); nrm2 += v * v; }
  float nrm = sqrtf(nrm2);
  if (nrm == 0.0f) nrm = sqrtf((float)Tn);
  const float inv = 1.0f / nrm;
  for (int i = 0; i < Tn; ++i) Dg[i * Kn + k] = dict_elem(k, i, Drr, Dth) * inv;
}

// --------------------------------------------- var = softplus(x^T W + b), (B,K,N)
__global__ __launch_bounds__(256) void k_var(const float* __restrict__ x,
                                             const float* __restrict__ W,
                                             const float* __restrict__ bvec,
                                             float* __restrict__ varout) {
  const size_t id = (size_t)blockIdx.x * 256 + threadIdx.x;
  if (id >= (size_t)Bn * Kn * Nn) return;
  const int n = (int)(id % Nn);
  const int k = (int)((id / Nn) % Kn);
  const int b = (int)(id / ((size_t)Nn * Kn));
  float acc = bvec[k];
  for (int t = 0; t < Tn; ++t)
    acc += x[((size_t)b * Tn + t) * Nn + n] * W[t * Kn + k];
  varout[id] = (acc > 20.0f) ? acc : log1pf(expf(acc));
}

// ------------------------------------------------- Dty[n,b,k] = sum_t D[t,k] x[b,t,n]
__global__ __launch_bounds__(256) void k_dty(const float* __restrict__ x,
                                             const float* __restrict__ Dg,
                                             float* __restrict__ dtyout) {
  const size_t id = (size_t)blockIdx.x * 256 + threadIdx.x;
  if (id >= (size_t)NB * Kn) return;
  const int k = (int)(id % Kn);
  const int b = (int)((id / Kn) % Bn);
  const int n = (int)(id / ((size_t)Kn * Bn));
  float acc = 0.0f;
  for (int t = 0; t < Tn; ++t)
    acc += Dg[t * Kn + k] * x[((size_t)b * Tn + t) * Nn + n];
  dtyout[id] = acc;
}

// -------- M = D diag(v) D^T + I/beta via WMMA: (48x176)*(176x48), 44 k-steps
__global__ __launch_bounds__(256) void k_m(const float* __restrict__ Dg,
                                           const float* __restrict__ varg,
                                           const float* __restrict__ logbeta,
                                           float* __restrict__ Mg) {
  __shared__ float Dsh[TP][KP];   // D zero-padded to 48 x 176
  __shared__ float vsh[KP];
  const int pair = blockIdx.x;
  const int n = pair / Bn, b = pair % Bn;
  const int tid = threadIdx.x;
  for (int idx = tid; idx < TP * KP; idx += 256) {
    const int t = idx / KP, k = idx % KP;
    Dsh[t][k] = (t < Tn && k < Kn) ? Dg[t * Kn + k] : 0.0f;
  }
  for (int k = tid; k < KP; k += 256)
    vsh[k] = (k < Kn) ? varg[((size_t)b * Kn + k) * Nn + n] : 0.0f;
  __syncthreads();
  const float invbeta = expf(-logbeta[0]);

  const int wave = tid >> 5, lane = tid & 31;
  const int lm = lane & 15, lh = lane >> 4;

  for (int tile = wave; tile < MT * MT; tile += 8) {   // 9 output tiles
    const int mt = tile / MT, st = tile % MT;
    const int mbase = mt * 16, sbase = st * 16;
    v8f c = {};
#pragma unroll
    for (int step = 0; step < KP / 4; ++step) {        // 44 WMMA steps
      const int k0 = step * 4 + lh * 2;
      v2f a, bb;
      a.x  = Dsh[mbase + lm][k0];                      // A[t][k] = D[t,k]
      a.y  = Dsh[mbase + lm][k0 + 1];
      bb.x = vsh[k0]     * Dsh[sbase + lm][k0];        // B[k][s] = v[k] D[s,k]
      bb.y = vsh[k0 + 1] * Dsh[sbase + lm][k0 + 1];
      c = __builtin_amdgcn_wmma_f32_16x16x4_f32(false, a, false, bb,
                                                (short)0, c, false, false);
    }
    const int s = sbase + lm;
#pragma unroll
    for (int i = 0; i < 8; ++i) {
      const int t = mbase + i + lh * 8;
      if (t < Tn && s < Tn)
        Mg[(size_t)pair * Tn * Tn + t * Tn + s] = c[i] + ((t == s) ? invbeta : 0.0f);
    }
  }
}

// -------------------------------- S = inv(M), Gauss-Jordan (SPD, no pivoting)
__global__ __launch_bounds__(64) void k_inv(const float* __restrict__ Mg,
                                            float* __restrict__ Sg) {
  __shared__ float A[Tn][2 * Tn + 1];
  __shared__ float s_pinv;
  const int pair = blockIdx.x, tid = threadIdx.x;
  for (int idx = tid; idx < Tn * Tn; idx += 64) {
    const int t = idx / Tn, s = idx % Tn;
    A[t][s]      = __builtin_nontemporal_load(&Mg[(size_t)pair * Tn * Tn + idx]);
    A[t][Tn + s] = (t == s) ? 1.0f : 0.0f;
  }
  __syncthreads();
  for (int p = 0; p < Tn; ++p) {
    if (tid == 0) s_pinv = 1.0f / A[p][p];
    __syncthreads();
    const float pinv = s_pinv;
    for (int c = tid; c < 2 * Tn; c += 64) A[p][c] *= pinv;
    __syncthreads();
    if (tid < Tn && tid != p) {
      const float f = A[tid][p];
      for (int c = 0; c < 2 * Tn; ++c) A[tid][c] -= f * A[p][c];
    }
    __syncthreads();
  }
  for (int idx = tid; idx < Tn * Tn; idx += 64) {
    const int t = idx / Tn, s = idx % Tn;
    Sg[(size_t)pair * Tn * Tn + idx] = A[t][Tn + s];
  }
}

// ------- Sigma/mu/loss: E = S D and Q = D^T E both via V_WMMA_F32_16X16X4_F32,
//         R = (v v^T) .* Q, Sigma = diag(v) - R (non-temporal stream-out),
//         mu = beta * Sigma * Dty fused, loss partial per pair
__global__ __launch_bounds__(256) void k_sigma(const float* __restrict__ Dg,
                                               const float* __restrict__ Sg,
                                               const float* __restrict__ varg,
                                               const float* __restrict__ dtyg,
                                               const float* __restrict__ logbeta,
                                               float* __restrict__ out_mu,
                                               float* __restrict__ out_sig,
                                               float* __restrict__ pairloss) {
  __shared__ float Dsh[Tn][KP];       // D, zero-padded cols
  __shared__ float Esh[Tn][KP];       // E = S * D
  __shared__ float Ssh[TP][Tn + 4];   // S, zero-padded to 48 rows
  __shared__ float vsh[KP];
  __shared__ float dtysh[KP];
  __shared__ float mush[KP];
  __shared__ float qdiag[KP];
  __shared__ float redsh[8];

  const int pair = blockIdx.x;
  const int n = pair / Bn, b = pair % Bn;
  const int tid = threadIdx.x;
  const float beta = expf(logbeta[0]);

  for (int idx = tid; idx < Tn * KP; idx += 256) {
    const int t = idx / KP, k = idx % KP;
    Dsh[t][k] = (k < Kn) ? Dg[t * Kn + k] : 0.0f;
  }
  for (int idx = tid; idx < TP * Tn; idx += 256) {
    const int t = idx / Tn, s = idx % Tn;
    Ssh[t][s] = (t < Tn)
        ? __builtin_nontemporal_load(&Sg[(size_t)pair * Tn * Tn + idx])
        : 0.0f;
  }
  for (int k = tid; k < KP; k += 256) {
    vsh[k]   = (k < Kn) ? varg[((size_t)b * Kn + k) * Nn + n] : 0.0f;
    dtysh[k] = (k < Kn) ? dtyg[(size_t)pair * Kn + k] : 0.0f;
    qdiag[k] = 0.0f;
    mush[k]  = 0.0f;
  }
  __syncthreads();

  const int wave = tid >> 5, lane = tid & 31;
  const int lm = lane & 15, lh = lane >> 4;   // half-wave select

  // ---- E = S @ D via WMMA: 3 x 11 tiles, 9 k-steps each
  for (int tile = wave; tile < MT * NT; tile += 8) {
    const int mt = tile / NT, jt = tile % NT;
    const int mbase = mt * 16, jbase = jt * 16;
    v8f c = {};
#pragma unroll
    for (int step = 0; step < Tn / 4; ++step) {
      const int s0 = step * 4 + lh * 2;
      v2f a, bb;
      a.x  = Ssh[mbase + lm][s0];               // A[t][s] = S[t,s]
      a.y  = Ssh[mbase + lm][s0 + 1];
      bb.x = Dsh[s0    ][jbase + lm];           // B[s][j] = D[s,j]
      bb.y = Dsh[s0 + 1][jbase + lm];
      c = __builtin_amdgcn_wmma_f32_16x16x4_f32(false, a, false, bb,
                                                (short)0, c, false, false);
    }
    const int j = jbase + lm;
#pragma unroll
    for (int i = 0; i < 8; ++i) {
      const int t = mbase + i + lh * 8;
      if (t < Tn) Esh[t][j] = c[i];
    }
  }
  __syncthreads();

  // ---- Q = D^T @ E; each wave owns whole k-tile rows (deterministic mu)
  for (int kt = wave; kt < NT; kt += 8) {
    const int kbase = kt * 16;
    float rowacc[8];
#pragma unroll
    for (int i = 0; i < 8; ++i) rowacc[i] = 0.0f;

    for (int jt = 0; jt < NT; ++jt) {
      const int jbase = jt * 16;
      v8f c = {};
#pragma unroll
      for (int step = 0; step < Tn / 4; ++step) {   // 9 WMMA steps
        const int t0 = step * 4 + lh * 2;
        v2f a, bb;
        a.x  = Dsh[t0    ][kbase + lm];             // A = D^T tile (16x4)
        a.y  = Dsh[t0 + 1][kbase + lm];
        bb.x = Esh[t0    ][jbase + lm];             // B = E tile (4x16)
        bb.y = Esh[t0 + 1][jbase + lm];
        c = __builtin_amdgcn_wmma_f32_16x16x4_f32(false, a, false, bb,
                                                  (short)0, c, false, false);
      }
      const int j = jbase + lm;
      const bool jv = (j < Kn);
      const float vj = vsh[j], dtj = dtysh[j];
      // one per-lane base pointer; row stride K*4 B goes in the store ioffset
      float* rowp = out_sig + ((size_t)pair * Kn + (size_t)(kbase + lh * 8)) * Kn + j;
#pragma unroll
      for (int i = 0; i < 8; ++i) {                 // C/D: rows i (+8 for hi half)
        const int k = kbase + i + lh * 8;
        const float q = c[i];
        const float sig = ((k == j) ? vsh[k] : 0.0f) - vsh[k] * vj * q;
        if (jv && k < Kn) {
          __builtin_nontemporal_store(sig, rowp + i * Kn);  // 212 MB stream, skip L2 residency
          if (k == j) qdiag[k] = q;
        }
        rowacc[i] += sig * dtj;                     // padded j contributes 0
      }
    }
#pragma unroll
    for (int i = 0; i < 8; ++i) {                   // reduce over 16 j-lanes
      float s = rowacc[i];
      s += __shfl_xor(s, 1, 32);
      s += __shfl_xor(s, 2, 32);
      s += __shfl_xor(s, 4, 32);
      s += __shfl_xor(s, 8, 32);
      if (lm == 0) mush[kbase + i + lh * 8] = beta * s;
    }
  }
  __syncthreads();

  // mu out + loss partial: (mu^2 - R_kk)^2, R_kk = v_k^2 Q_kk
  float lpart = 0.0f;
  for (int k = tid; k < Kn; k += 256) {
    const float mu = mush[k];
    out_mu[(size_t)pair * Kn + k] = mu;
    const float rkk = vsh[k] * vsh[k] * qdiag[k];
    const float d = mu * mu - rkk;
    lpart += d * d;
  }
#pragma unroll
  for (int off = 16; off > 0; off >>= 1) lpart += __shfl_xor(lpart, off, 32);
  if (lane == 0) redsh[wave] = lpart;
  __syncthreads();
  if (tid == 0) {
    float s = 0.0f;
#pragma unroll
    for (int w = 0; w < 8; ++w) s += redsh[w];
    pairloss[pair] = s;                             // deterministic block sum
  }
}

// ------------------------------------------------------------- loss finalize
__global__ void k_loss(const float* __restrict__ pl, float* __restrict__ out_loss) {
  if (threadIdx.x == 0 && blockIdx.x == 0) {
    float s = 0.0f;
    for (int i = 0; i < NB; ++i) s += pl[i];
    out_loss[0] = s / (float)((size_t)NB * Kn);
  }
}

extern "C" void kernel_launch(void* const* d_in, const int* in_sizes, int n_in,
                              void* d_out, int out_size, void* d_ws, size_t ws_size,
                              hipStream_t stream) {
  (void)in_sizes; (void)n_in; (void)out_size; (void)ws_size;
  const float* x   = (const float*)d_in[0];
  const float* Drr = (const float*)d_in[1];
  const float* Dth = (const float*)d_in[2];
  const float* W   = (const float*)d_in[3];
  const float* ba  = (const float*)d_in[4];
  const float* lb  = (const float*)d_in[5];
  float* out = (float*)d_out;
  float* ws  = (float*)d_ws;

  float* Dg  = ws + WS_D;
  float* dty = ws + WS_DTY;
  float* Mg  = ws + WS_M;
  float* Sg  = ws + WS_S;
  float* pl  = ws + WS_PL;

  float* out_mu  = out + OFF_MU;
  float* out_sig = out + OFF_SIGMA;
  float* out_ls  = out + OFF_LOSS;
  float* out_var = out + OFF_VAR;

  k_dict<<<1, 192, 0, stream>>>(Drr, Dth, Dg);
  k_var<<<((size_t)Bn * Kn * Nn + 255) / 256, 256, 0, stream>>>(x, W, ba, out_var);
  k_dty<<<((size_t)NB * Kn + 255) / 256, 256, 0, stream>>>(x, Dg, dty);
  k_m<<<NB, 256, 0, stream>>>(Dg, out_var, lb, Mg);
  k_inv<<<NB, 64, 0, stream>>>(Mg, Sg);
  k_sigma<<<NB, 256, 0, stream>>>(Dg, Sg, out_var, dty, lb, out_mu, out_sig, pl);
  k_loss<<<1, 32, 0, stream>>>(pl, out_ls);
}